// DWT_5540507811878
// MI455X (gfx1250) — compile-verified
//
#include <hip/hip_runtime.h>

// 2D Haar DWT, (8,512,512,32) f32 -> (8,256,256,128) f32 [ll|lh|hl|hh on C].
// Memory-bound streaming kernel: async global->LDS copies (ASYNCcnt pipeline),
// butterfly gather done on the LDS read side, non-temporal b128 stores.

typedef float v4f __attribute__((ext_vector_type(4)));

#define GROUPS_PER_WAVE 16u   // 16 groups of 4 output pixels per wave (same row)

__global__ __launch_bounds__(256) void haar_dwt_kernel(const float* __restrict__ in,
                                                       float* __restrict__ out) {
    // 8 waves * 2 buffers * 128 float4 (2KB) = 32 KB
    __shared__ v4f sm[2048];

    const unsigned lane = threadIdx.x & 31u;
    const unsigned warp = threadIdx.x >> 5;
    const unsigned wave_global = blockIdx.x * 8u + warp;

    // One "group" = 4 consecutive output pixels (needs 2 input rows x 8 pixels).
    // groups per output row = 256/4 = 64 ; per batch image = 256*64 = 16384.
    const unsigned g0  = wave_global * GROUPS_PER_WAVE;   // < 131072
    const unsigned b   = g0 >> 14;
    const unsigned rem = g0 & 16383u;
    const unsigned u   = rem >> 6;     // output row
    const unsigned tg0 = rem & 63u;    // first group in row (16 | 64 -> same row)

    // Global input byte address of this wave's first group, lane-resolved.
    unsigned long long p0 = (unsigned long long)((const char*)in
        + (size_t)b   * 33554432u     // 512*512*32*4
        + (size_t)u   * 131072u       // two input rows per output row
        + (size_t)tg0 * 1024u         // 8 input pixels * 128 B per group
        + (size_t)lane * 16u);

    const unsigned sm4 = warp * 256u;  // per-wave float4 index base
    const unsigned c4  = lane & 7u;    // channel chunk (4 channels)
    const unsigned pp  = lane >> 3;    // output pixel within group (0..3)

    unsigned long long opix =
        ((unsigned long long)(b * 256u + u) * 256u + tg0 * 4u + pp);
    v4f* out4 = (v4f*)out;

    // Issue the 4 async copies for one group into LDS buffer `par`.
    // LDS address operand is derived from &sm[...] (ptrtoint): the low 32 bits
    // of the flat shared pointer are the LDS byte offset, and the capture
    // forces the compiler to treat the asm as writing sm.
    auto issue = [&](unsigned long long q0, unsigned par) {
        unsigned l0 = (unsigned)(size_t)&sm[sm4 + par * 128u + lane]; // row 2u half
        unsigned l1 = l0 + 1024u;                                     // row 2u+1 half
        unsigned long long q1 = q0 + 65536u;                          // next input row
        asm volatile("global_load_async_to_lds_b128 %0, %1, off th:TH_LOAD_NT"
                     :: "v"(l0), "v"(q0) : "memory");
        asm volatile("global_load_async_to_lds_b128 %0, %1, off offset:512 th:TH_LOAD_NT"
                     :: "v"(l0), "v"(q0) : "memory");
        asm volatile("global_load_async_to_lds_b128 %0, %1, off th:TH_LOAD_NT"
                     :: "v"(l1), "v"(q1) : "memory");
        asm volatile("global_load_async_to_lds_b128 %0, %1, off offset:512 th:TH_LOAD_NT"
                     :: "v"(l1), "v"(q1) : "memory");
    };

    issue(p0, 0u);
    for (unsigned i = 0; i < GROUPS_PER_WAVE; ++i) {
        const unsigned par = i & 1u;
        if (i + 1u < GROUPS_PER_WAVE) {
            // LDS reads of the buffer we are about to refill finished long ago.
            asm volatile("s_wait_dscnt 0" ::: "memory");
            issue(p0 + (unsigned long long)(i + 1u) * 1024u, (i + 1u) & 1u);
            // 4 newest copies may be pending; previous 4 (our buffer) are done.
            asm volatile("s_wait_asynccnt 4" ::: "memory");
        } else {
            asm volatile("s_wait_asynccnt 0" ::: "memory");
        }

        const unsigned base = sm4 + par * 128u + pp * 16u + c4;
        v4f a  = sm[base];        // x00: row 2u,  even pixel
        v4f bb = sm[base + 8];    // x01: row 2u,  odd  pixel
        v4f c  = sm[base + 64];   // x10: row 2u+1, even pixel
        v4f d  = sm[base + 72];   // x11: row 2u+1, odd  pixel

        v4f s0 = a + bb, s1 = c + d;      // row sums   (lowpass W)
        v4f d0 = bb - a, d1 = d - c;      // row diffs  (highpass W)
        v4f ll = 0.5f * (s0 + s1);
        v4f lh = 0.5f * (s1 - s0);
        v4f hl = 0.5f * (d0 + d1);
        v4f hh = 0.5f * (d1 - d0);

        unsigned long long ob = (opix + (unsigned long long)i * 4u) * 32u + c4;
        __builtin_nontemporal_store(ll, &out4[ob]);
        __builtin_nontemporal_store(lh, &out4[ob + 8]);
        __builtin_nontemporal_store(hl, &out4[ob + 16]);
        __builtin_nontemporal_store(hh, &out4[ob + 24]);
    }
}

extern "C" void kernel_launch(void* const* d_in, const int* in_sizes, int n_in,
                              void* d_out, int out_size, void* d_ws, size_t ws_size,
                              hipStream_t stream) {
    (void)in_sizes; (void)n_in; (void)out_size; (void)d_ws; (void)ws_size;
    const float* x = (const float*)d_in[0];   // (8,512,512,32) f32
    float* out = (float*)d_out;               // (8,256,256,128) f32
    // 131072 groups / 16 per wave = 8192 waves / 8 waves per block = 1024 blocks
    haar_dwt_kernel<<<1024, 256, 0, stream>>>(x, out);
}